// HaarDWT_77506979824171
// MI455X (gfx1250) — compile-verified
//
#include <hip/hip_runtime.h>

typedef __attribute__((ext_vector_type(4))) float v4f;

// Geometry fixed by the reference: x (4, 64, 512, 512) fp32 -> y (4, 256, 256, 256).
//   planes P = 256, input plane 512*512 floats, output plane 256*256 floats,
//   out channels per plane p: [4p..4p+3] = ll, lh, hl, hh.
//
// Tile = half an output row of one plane = 2 input rows x 256 floats (2 KB).
// One wave (32 lanes) owns a tile: lane l computes output pixels [4l, 4l+4).
// Tiles per wave: NITER, double-buffered through LDS via
// GLOBAL_LOAD_ASYNC_TO_LDS_B128 (ASYNCcnt), no workgroup barriers needed.

#define WAVES_PER_BLOCK 8u
#define NITER           8u

// Low 32 bits of a generic pointer into __shared__ = wave-relative LDS byte
// offset (generic LDS addr = {SHARED_BASE aperture, offset[31:0]}).
__device__ __forceinline__ unsigned lds_addr32(const void* p) {
    return (unsigned)(unsigned long long)p;
}

__global__ __launch_bounds__(256) void haar_dwt_async(const float* __restrict__ x,
                                                      float* __restrict__ y) {
    constexpr unsigned W    = 512u;           // input width (floats)
    constexpr unsigned INP  = 512u * 512u;    // input plane elems
    constexpr unsigned OUTP = 256u * 256u;    // output plane elems

    // 2 buffers x 512 floats per wave = 4 KB/wave, 32 KB/block.
    __shared__ float lds[WAVES_PER_BLOCK * 1024u];

    const unsigned lane  = threadIdx.x & 31u;
    const unsigned waveL = threadIdx.x >> 5;
    const unsigned w     = blockIdx.x * WAVES_PER_BLOCK + waveL;

    float* wave_lds = &lds[waveL * 1024u];

    // Issue 4 async b128 copies: tile T -> LDS buffer `buf`.
    // Lane l copies bytes [16l,16l+16) of each 512B half; offsets {0,512}
    // cover a 1024B input row (the instruction offset is added to BOTH the
    // LDS and the global address, so row0/row1 get separate base registers).
    auto issue = [&](unsigned T, unsigned buf) {
        const unsigned p    = T >> 9;
        const unsigned r    = (T >> 1) & 255u;
        const unsigned half = T & 1u;
        const float* src = x + (size_t)p * INP + (size_t)(2u * r) * W + (size_t)(half * 256u);
        const float* g0 = src + 4u * lane;      // row 2r,   16B per lane
        const float* g1 = g0 + W;               // row 2r+1
        unsigned l0 = lds_addr32(wave_lds + buf * 512u) + 16u * lane;
        unsigned l1 = l0 + 1024u;
        asm volatile(
            "global_load_async_to_lds_b128 %0, %2, off\n\t"
            "global_load_async_to_lds_b128 %0, %2, off offset:512\n\t"
            "global_load_async_to_lds_b128 %1, %3, off\n\t"
            "global_load_async_to_lds_b128 %1, %3, off offset:512\n\t"
            "s_wait_xcnt 0x0"
            :: "v"(l0), "v"(l1), "v"(g0), "v"(g1)
            : "memory");
    };

    issue(w * NITER, 0u);   // prime the pipeline

    for (unsigned i = 0; i < NITER; ++i) {
        const unsigned T   = w * NITER + i;
        const unsigned buf = i & 1u;

        if (i + 1u < NITER) {
            issue(T + 1u, buf ^ 1u);   // 8 outstanding after this
            asm volatile("s_wait_asynccnt 0x4" ::: "memory");  // drain tile i's 4
        } else {
            asm volatile("s_wait_asynccnt 0x0" ::: "memory");
        }

        // Consume tile i from LDS: lane l reads floats [8l,8l+8) of each row.
        const float* sbuf = wave_lds + buf * 512u;
        const v4f* s0 = (const v4f*)(sbuf + 8u * lane);           // row even
        const v4f* s1 = (const v4f*)(sbuf + 256u + 8u * lane);    // row odd
        v4f r0a = s0[0], r0b = s0[1];
        v4f r1a = s1[0], r1b = s1[1];

        // De-interleave even/odd columns for 4 output pixels.
        v4f a = { r0a.x, r0a.z, r0b.x, r0b.z };
        v4f b = { r0a.y, r0a.w, r0b.y, r0b.w };
        v4f c = { r1a.x, r1a.z, r1b.x, r1b.z };
        v4f d = { r1a.y, r1a.w, r1b.y, r1b.w };

        // Haar butterfly.
        v4f s0v = a + b, d0v = a - b;
        v4f s1v = c + d, d1v = c - d;
        v4f ll = (s0v + s1v) * 0.5f;
        v4f lh = (d0v + d1v) * 0.5f;
        v4f hl = (s0v - s1v) * 0.5f;
        v4f hh = (d0v - d1v) * 0.5f;

        const unsigned p    = T >> 9;
        const unsigned r    = (T >> 1) & 255u;
        const unsigned half = T & 1u;
        float* dst = y + (size_t)p * (4u * OUTP) + (size_t)r * 256u
                       + (size_t)(half * 128u) + (size_t)(4u * lane);

        __builtin_nontemporal_store(ll, (v4f*)(dst));
        __builtin_nontemporal_store(lh, (v4f*)(dst + OUTP));
        __builtin_nontemporal_store(hl, (v4f*)(dst + 2u * OUTP));
        __builtin_nontemporal_store(hh, (v4f*)(dst + 3u * OUTP));
    }
}

extern "C" void kernel_launch(void* const* d_in, const int* in_sizes, int n_in,
                              void* d_out, int out_size, void* d_ws, size_t ws_size,
                              hipStream_t stream) {
    (void)in_sizes; (void)n_in; (void)out_size; (void)d_ws; (void)ws_size;
    const float* x = (const float*)d_in[0];
    float*       y = (float*)d_out;

    // tiles = 256 planes * 256 rows * 2 halves = 131072
    // waves = 131072 / NITER = 16384 ; blocks = 16384 / 8 = 2048
    constexpr unsigned blocks = (256u * 256u * 2u) / NITER / WAVES_PER_BLOCK;

    haar_dwt_async<<<blocks, 256, 0, stream>>>(x, y);
}